// RModel_82806969466959
// MI455X (gfx1250) — compile-verified
//
#include <hip/hip_runtime.h>

// RModel rollout: B=16384 samples, H=1024, NMAX=10.
// One wave32 handles 16 samples. Hidden layer = V_WMMA_F32_16X16X4_F32 with
// bias folded via K=4 trick: A rows = [x, y, 1, 0], B rows = [W1x, W1y, b1, 0].
// Output layer (H -> 1) = per-lane fma against W2 + LDS 16x16 transpose reduce.

#define HDIM  1024
#define NT    (HDIM / 16)   // 64 WMMA tiles over the hidden dim
#define NMAXI 10
#define WAVES 4
#define BLOCK (WAVES * 32)

typedef __attribute__((ext_vector_type(2))) float v2f;
typedef __attribute__((ext_vector_type(4))) float v4f;
typedef __attribute__((ext_vector_type(8))) float v8f;

struct SharedBlob {
    // B-operand tiles: [net][tile][lane] -> (K-row pair per ISA layout)
    //   lanes 0..15 : (W1[j][0], W1[j][1])   (K=0,1 rows)
    //   lanes 16..31: (b1[j],    0.0f)       (K=2,3 rows)
    v2f   bop[2][NT][32];
    float w2[2][HDIM];
    float xp[WAVES][16][16];  // per-wave transpose pad for the over-N reduce
};

__device__ __forceinline__ float elu_f(float z) {
    return z > 0.0f ? z : (__expf(z) - 1.0f);
}

// Evaluate one MLP (2 -> H -> 1) for the wave's 16 samples.
// Every lane returns the output of sample (lane & 15).
__device__ __forceinline__ float mlp16(const v2f* __restrict__ bop,  // [NT*32]
                                       const float* __restrict__ w2, // [HDIM]
                                       float* __restrict__ xp,       // this wave's 16x16
                                       int lane, float ax, float ay, float b2)
{
    v2f a;
    a.x = ax;
    a.y = ay;

    float acc[8];
#pragma unroll
    for (int r = 0; r < 8; ++r) acc[r] = 0.0f;

    const int n  = lane & 15;
    const int hi = lane >> 4;

#pragma unroll 4
    for (int t = 0; t < NT; ++t) {
        v2f b   = bop[t * 32 + lane];   // ds_load_b64
        float w = w2[t * 16 + n];       // ds_load_b32
        v8f c = {};
        // D[M=sample, N=hidden] = A(16x4) x B(4x16): x*W1x + y*W1y + 1*b1 + 0
        v8f d = __builtin_amdgcn_wmma_f32_16x16x4_f32(
            /*neg_a=*/false, a, /*neg_b=*/false, b,
            /*c_mod=*/(short)0, c, /*reuse_a=*/false, /*reuse_b=*/false);
#pragma unroll
        for (int r = 0; r < 8; ++r)
            acc[r] = fmaf(elu_f(d[r]), w, acc[r]);
    }

    // Transpose-reduce over N via LDS: lane holds (M = r + 8*hi, N = lane&15).
#pragma unroll
    for (int r = 0; r < 8; ++r)
        xp[(r + 8 * hi) * 16 + n] = acc[r];   // ds_store_b32 x8
    asm volatile("s_wait_dscnt 0x0" ::: "memory");

    const v4f* row = (const v4f*)(xp + n * 16);  // ds_load_b128 x4
    v4f s4 = row[0] + row[1] + row[2] + row[3];
    float s = (s4.x + s4.y) + (s4.z + s4.w);
    asm volatile("" ::: "memory");  // keep later xp stores from hoisting over reads
    return s + b2;
}

__global__ __launch_bounds__(BLOCK)
void rmodel_rollout_kernel(const float* __restrict__ X,
                           const float* __restrict__ Y,
                           const float* __restrict__ deW1, const float* __restrict__ deB1,
                           const float* __restrict__ deW2, const float* __restrict__ deB2,
                           const float* __restrict__ vaW1, const float* __restrict__ vaB1,
                           const float* __restrict__ vaW2, const float* __restrict__ vaB2,
                           float* __restrict__ out, int Bn)
{
    __shared__ SharedBlob sm;

    const int tid  = threadIdx.x;
    const int lane = tid & 31;
    const int wv   = tid >> 5;

    // ---- Stage weights into LDS (once per block) ----
    for (int idx = tid; idx < 2 * NT * 32; idx += BLOCK) {
        const int net = idx / (NT * 32);
        const int rem = idx % (NT * 32);
        const int t = rem >> 5, l = rem & 31;
        const float* W1 = net ? vaW1 : deW1;
        const float* B1 = net ? vaB1 : deB1;
        const int j = t * 16 + (l & 15);
        v2f b;
        if (l < 16) { b.x = W1[2 * j]; b.y = W1[2 * j + 1]; }
        else        { b.x = B1[j];     b.y = 0.0f;          }
        sm.bop[net][t][l] = b;
    }
    for (int idx = tid; idx < 2 * HDIM; idx += BLOCK) {
        const int net = idx / HDIM, j = idx % HDIM;
        sm.w2[net][j] = net ? vaW2[j] : deW2[j];
    }
    __syncthreads();

    const float de_b2 = deB2[0];
    const float va_b2 = vaB2[0];

    const int m     = lane & 15;
    const int s     = (blockIdx.x * WAVES + wv) * 16 + m;
    const bool loA  = (lane < 16);
    const bool live = loA && (s < Bn);

    // All lanes mirror sample (lane & 15)'s state so EXEC stays all-1s.
    float x = (s < Bn) ? X[s] : 0.0f;
    float y = (s < Bn) ? Y[s] : 0.0f;

    const v2f* deBop = &sm.bop[0][0][0];
    const v2f* vaBop = &sm.bop[1][0][0];
    float* xp = &sm.xp[wv][0][0];

    const float ax  = loA ? x : 1.0f;   // lanes 16..31 supply K=2 row constant 1 (bias)
    const int  offV = (NMAXI + 1) * Bn; // v_stack after y_stack in flat output

    // v0 = val_mlp([x, Y])
    float v0 = mlp16(vaBop, sm.w2[1], xp, lane, ax, loA ? y : 0.0f, va_b2);
    if (live) { out[s] = y; out[offV + s] = v0; }

    float v = v0;
    bool done = false;
#pragma unroll 1
    for (int i = 0; i < NMAXI; ++i) {
        float de    = mlp16(deBop, sm.w2[0], xp, lane, ax, loA ? y : 0.0f, de_b2);
        float y_new = y + de;
        float v_new = mlp16(vaBop, sm.w2[1], xp, lane, ax, loA ? y_new : 0.0f, va_b2);

        float y_out = done ? y : y_new;
        float v_out = done ? v : v_new;
        if (live) {
            out[(i + 1) * Bn + s]        = y_out;
            out[offV + (i + 1) * Bn + s] = v_out;
        }
        done = done || ((i > 0) && (v_new >= v));  // compare against frozen v_prev
        y = y_out;
        v = v_out;
    }
}

extern "C" void kernel_launch(void* const* d_in, const int* in_sizes, int n_in,
                              void* d_out, int out_size, void* d_ws, size_t ws_size,
                              hipStream_t stream) {
    const float* X    = (const float*)d_in[0];
    const float* Y    = (const float*)d_in[1];
    const float* deW1 = (const float*)d_in[2];
    const float* deB1 = (const float*)d_in[3];
    const float* deW2 = (const float*)d_in[4];
    const float* deB2 = (const float*)d_in[5];
    const float* vaW1 = (const float*)d_in[6];
    const float* vaB1 = (const float*)d_in[7];
    const float* vaW2 = (const float*)d_in[8];
    const float* vaB2 = (const float*)d_in[9];
    float* out = (float*)d_out;

    const int Bn = in_sizes[0];                       // batch size (X flat count)
    const int samples_per_block = WAVES * 16;         // 64
    const int blocks = (Bn + samples_per_block - 1) / samples_per_block;

    rmodel_rollout_kernel<<<blocks, BLOCK, 0, stream>>>(
        X, Y, deW1, deB1, deW2, deB2, vaW1, vaB1, vaW2, vaB2, out, Bn);
}